// SimpleFeedback_10024453669550
// MI455X (gfx1250) — compile-verified
//
#include <hip/hip_runtime.h>

// CDNA5 (gfx1250) wave32 kernel: batched Tsit5 integration of damped point mass
// with MLP feedback controller. One wave owns 16 batch rows for all 100 steps.
//
// Layer 1 ([16,6] x [6,128]) -> 16x V_WMMA_F32_16X16X4_F32 per stage (K=6 padded
// to 8), with W1/b1 pre-scaled by 2*log2(e) so the WMMA output feeds v_exp_f32
// directly. tanh's affine tail (1 - 2r) is folded into layer 2:
//   sum_n tanh_n*W2[c,n] = sum_n W2[c,n] - 2*sum_n r_n*W2[c,n],
// so the partials seed with sum(W2) and accumulate r * (-2*W2).
// Layer 2 reduction: DPP16 row-rotation butterfly (pure VALU) + 32-float LDS
// bounce to scatter the 2 controls back to the A-fragment lanes.

typedef __attribute__((ext_vector_type(2))) float v2f;
typedef __attribute__((ext_vector_type(8))) float v8f;

#define B_TOTAL 4096
#define HIDDEN  128
#define N_STEPS 100
#define DT      0.01f
#define DAMPING 0.1f

// x += row_ror:N(x)  (DPP16: rotation within each 16-lane row; pure VALU).
// Applying N = 8,4,2,1 gives every lane the full 16-lane row sum.
#define DPP_ROR_ADD(x, N)                                                     \
    x += __int_as_float(__builtin_amdgcn_update_dpp(                          \
            0, __float_as_int(x), 0x120 | (N), 0xf, 0xf, true))

__global__ __launch_bounds__(32)
void tsit5_feedback_kernel(const float* __restrict__ state,
                           const float* __restrict__ inputs,
                           const float* __restrict__ W1,
                           const float* __restrict__ b1,
                           const float* __restrict__ W2,
                           const float* __restrict__ b2,
                           float* __restrict__ out)
{
    __shared__ float ctrlLDS[32];   // [c=0..1][row=0..15]

    const int  lane  = threadIdx.x & 31;
    const int  nlo   = lane & 15;
    const bool lower = lane < 16;              // lanes 0-15: comps 0,1 ; 16-31: comps 2,3
    const int  gRow  = blockIdx.x * 16 + nlo;  // batch row this lane covers

    const float SCL = 2.8853900817779268f;     // 2*log2(e): tanh(x)=1-2/(exp2(SCL*x)+1)

    // ---- Preload weights into registers (constant across all 600 stages) ----
    // B-fragment layout for V_WMMA_F32_16X16X4_F32 (B is KxN = 4x16):
    //   vgpr0: lanes 0-15 -> K=0, lanes 16-31 -> K=2 ; vgpr1: K=1 / K=3, N = lane%16.
    // GEMM1 computes net_in @ (SCL*W1)^T, so B[k][n] = SCL*W1[n][k], n = t*16+nlo.
    v2f   bf0[8], bf1[8];            // K-chunk 0 (net_in cols 0-3), chunk 1 (cols 4-5, pad 6-7=0)
    float b1v[8], w2m0[8], w2m1[8];  // scaled bias; -2*W2 rows
    float W2s0 = 0.0f, W2s1 = 0.0f;  // per-lane partial of sum_n W2[c,n]
#pragma unroll
    for (int t = 0; t < 8; ++t) {
        const int n = t * 16 + nlo;
        bf0[t].x = SCL * W1[n * 6 + (lower ? 0 : 2)];
        bf0[t].y = SCL * W1[n * 6 + (lower ? 1 : 3)];
        bf1[t].x = lower ? SCL * W1[n * 6 + 4] : 0.0f;   // K=4 / K=6(pad)
        bf1[t].y = lower ? SCL * W1[n * 6 + 5] : 0.0f;   // K=5 / K=7(pad)
        b1v[t]   = SCL * b1[n];
        const float w20 = W2[0 * HIDDEN + n];
        const float w21 = W2[1 * HIDDEN + n];
        W2s0 += w20;          W2s1 += w21;
        w2m0[t] = -2.0f * w20; w2m1[t] = -2.0f * w21;
    }
    const float b2v0 = b2[0], b2v1 = b2[1];

    // A-fragment for net_in columns 4,5 (the constant external inputs); zero on upper
    // half lanes (K=6,7 padding).
    v2f afragIn;
    afragIn.x = lower ? inputs[gRow * 2 + 0] : 0.0f;
    afragIn.y = lower ? inputs[gRow * 2 + 1] : 0.0f;

    // State in A-fragment layout: 2 components per lane.
    float y0 = state[gRow * 4 + (lower ? 0 : 2)];
    float y1 = state[gRow * 4 + (lower ? 1 : 3)];

    // Trajectory slot 0 = initial state.
    {
        float2 o; o.x = y0; o.y = y1;
        *(float2*)(out + (size_t)gRow * 4 + (lower ? 0 : 2)) = o;
    }

    // Tsit5 tableau
    static constexpr float Ac[6][5] = {
        {0.f, 0.f, 0.f, 0.f, 0.f},
        {0.161f, 0.f, 0.f, 0.f, 0.f},
        {-0.008480655492356989f, 0.335480655492357f, 0.f, 0.f, 0.f},
        {2.8971530571054935f, -6.359448489975075f, 4.3622954328695815f, 0.f, 0.f},
        {5.325864828439257f, -11.748883564062828f, 7.4955393428898365f, -0.09249506636175525f, 0.f},
        {5.86145544294642f, -12.92096931784711f, 8.159367898576159f, -0.071584973281401f, -0.028269050394068383f}
    };
    static constexpr float Bw[6] = {
        0.09646076681806523f, 0.01f, 0.4798896504144996f,
        1.379008574103742f, -3.290069515436081f, 2.324710524099774f
    };

    float k0s[6], k1s[6];

#pragma unroll 1
    for (int step = 0; step < N_STEPS; ++step) {
#pragma unroll
        for (int s = 0; s < 6; ++s) {
            // Stage input z = y + DT * sum_j a[s][j] * k_j   (per-lane, 2 comps)
            float z0 = y0, z1 = y1;
#pragma unroll
            for (int j = 0; j < 5; ++j) {
                if (j < s) {
                    z0 = fmaf(DT * Ac[s][j], k0s[j], z0);
                    z1 = fmaf(DT * Ac[s][j], k1s[j], z1);
                }
            }

            // ---- vector_field(z) ----
            // Layer 1 (pre-scaled): c = SCL*(net_in @ W1^T + b1), via 16x WMMA.
            v2f a0; a0.x = z0; a0.y = z1;   // A chunk 0 is exactly the state layout
            float p0[8], p1[8];             // layer-2 partials (seeded with sum W2)
#pragma unroll
            for (int v = 0; v < 8; ++v) { p0[v] = W2s0; p1[v] = W2s1; }

#pragma unroll
            for (int t = 0; t < 8; ++t) {
                v8f c;
#pragma unroll
                for (int v = 0; v < 8; ++v) c[v] = b1v[t];   // seed acc with scaled bias
                c = __builtin_amdgcn_wmma_f32_16x16x4_f32(
                        false, a0,      false, bf0[t], (short)0, c, false, false);
                c = __builtin_amdgcn_wmma_f32_16x16x4_f32(
                        false, afragIn, false, bf1[t], (short)0, c, false, false);
                // r = 1/(exp2(c)+1);  tanh = 1-2r folded into the W2 FMAs.
#pragma unroll
                for (int v = 0; v < 8; ++v) {
                    const float e = __builtin_amdgcn_exp2f(c[v]);
                    const float r = __builtin_amdgcn_rcpf(e + 1.0f);
                    p0[v] = fmaf(r, w2m0[t], p0[v]);
                    p1[v] = fmaf(r, w2m1[t], p1[v]);
                }
            }

            // Butterfly-reduce over each 16-lane row via DPP row rotations
            // (rows 0-7 live in lanes 0-15, rows 8-15 in lanes 16-31, per WMMA
            // C layout; DPP16 rows coincide with these halves).
#pragma unroll
            for (int v = 0; v < 8; ++v) {
                DPP_ROR_ADD(p0[v], 8); DPP_ROR_ADD(p0[v], 4);
                DPP_ROR_ADD(p0[v], 2); DPP_ROR_ADD(p0[v], 1);
                DPP_ROR_ADD(p1[v], 8); DPP_ROR_ADD(p1[v], 4);
                DPP_ROR_ADD(p1[v], 2); DPP_ROR_ADD(p1[v], 1);
            }

            // Scatter controls to A-layout lanes through a 32-float LDS bounce.
            if (nlo == 0) {                       // lanes 0 and 16
                const int rbase = lower ? 0 : 8;
#pragma unroll
                for (int v = 0; v < 8; ++v) {
                    ctrlLDS[rbase + v]      = p0[v];
                    ctrlLDS[16 + rbase + v] = p1[v];
                }
            }
            __syncthreads();
            const float c0 = ctrlLDS[nlo]      + b2v0;
            const float c1 = ctrlLDS[16 + nlo] + b2v1;
            __syncthreads();

            // k: lower lanes (pos comps) take velocity = other half's z;
            //    upper lanes (vel comps) take control - DAMPING*vel.
            const float zv0 = __shfl_xor(z0, 16, 32);
            const float zv1 = __shfl_xor(z1, 16, 32);
            k0s[s] = lower ? zv0 : fmaf(-DAMPING, z0, c0);
            k1s[s] = lower ? zv1 : fmaf(-DAMPING, z1, c1);
        }

        // y_{n+1} = y_n + DT * sum b_j k_j ; store trajectory slot step+1.
#pragma unroll
        for (int j = 0; j < 6; ++j) {
            y0 = fmaf(DT * Bw[j], k0s[j], y0);
            y1 = fmaf(DT * Bw[j], k1s[j], y1);
        }
        float2 o; o.x = y0; o.y = y1;
        *(float2*)(out + ((size_t)(step + 1) * B_TOTAL + gRow) * 4 + (lower ? 0 : 2)) = o;
    }
}

extern "C" void kernel_launch(void* const* d_in, const int* in_sizes, int n_in,
                              void* d_out, int out_size, void* d_ws, size_t ws_size,
                              hipStream_t stream) {
    (void)in_sizes; (void)n_in; (void)out_size; (void)d_ws; (void)ws_size;
    const float* state  = (const float*)d_in[0];
    const float* inputs = (const float*)d_in[1];
    const float* W1     = (const float*)d_in[2];
    const float* b1     = (const float*)d_in[3];
    const float* W2     = (const float*)d_in[4];
    const float* b2     = (const float*)d_in[5];
    float* out = (float*)d_out;

    dim3 grid(B_TOTAL / 16);   // 256 waves, one 16-row tile each
    dim3 block(32);            // one wave32 per block
    tsit5_feedback_kernel<<<grid, block, 0, stream>>>(state, inputs, W1, b1, W2, b2, out);
}